// UnionRGATLayer_74431783240015
// MI455X (gfx1250) — compile-verified
//
#include <hip/hip_runtime.h>
#include <hip/hip_bf16.h>
#include <math.h>

#define D 128
#define WPAD 132  // LDS row pitch (floats) to avoid bank conflicts

typedef __attribute__((ext_vector_type(2))) float v2f;
typedef __attribute__((ext_vector_type(8))) float v8f;

// ---------- float atomic max via sign-split int punning (exact, always compiles) ----------
__device__ __forceinline__ void atomicMaxFloat(float* addr, float v) {
    if (v >= 0.0f) {
        atomicMax((int*)addr, __float_as_int(v));
    } else {
        atomicMin((unsigned int*)addr, __float_as_uint(v));
    }
}

// ---------- 1) u_{s,d,r}[k] = sum_j attn_fc_w[j, {0,128,256}+k] * v[j] ----------
__global__ void k_reduce_u(const float* __restrict__ attn_fc_w,
                           const float* __restrict__ v,
                           float* __restrict__ u) {
    int k = threadIdx.x;  // 0..127
    float as = 0.f, ad = 0.f, ar = 0.f;
    for (int j = 0; j < D; ++j) {
        float vj = v[j];
        const float* row = attn_fc_w + (size_t)j * (3 * D);
        as += row[k] * vj;
        ad += row[D + k] * vj;
        ar += row[2 * D + k] * vj;
    }
    u[k] = as; u[D + k] = ad; u[2 * D + k] = ar;
}

// ---------- 2) per-node / per-relation scalars (one wave32 per row) ----------
__global__ void k_scalars(const float* __restrict__ h,
                          const float* __restrict__ emb_rel,
                          const float* __restrict__ u,
                          float* __restrict__ ps, float* __restrict__ pd,
                          float* __restrict__ pr, int N, int R) {
    int gid  = blockIdx.x * blockDim.x + threadIdx.x;
    int row  = gid >> 5;
    int lane = gid & 31;
    if (row >= N + R) return;
    const float* us = u;
    const float* ud = u + D;
    const float* ur = u + 2 * D;
    if (row < N) {
        const float* hp = h + (size_t)row * D;
        float as = 0.f, ad = 0.f;
        for (int k = lane; k < D; k += 32) {
            float x = hp[k];
            as += x * us[k];
            ad += x * ud[k];
        }
        for (int off = 16; off > 0; off >>= 1) {
            as += __shfl_down(as, off);
            ad += __shfl_down(ad, off);
        }
        if (lane == 0) { ps[row] = as; pd[row] = ad; }
    } else {
        int r = row - N;
        const float* ep = emb_rel + (size_t)r * D;
        float ar = 0.f;
        for (int k = lane; k < D; k += 32) ar += ep[k] * ur[k];
        for (int off = 16; off > 0; off >>= 1) ar += __shfl_down(ar, off);
        if (lane == 0) pr[r] = ar;
    }
}

// ---------- 3) init: agg = 0, m = -inf, denom = 0 (every launch; ws not re-poisoned) ----------
__global__ void k_init(float* __restrict__ agg, float* __restrict__ m,
                       float* __restrict__ denom, long long nd, int N) {
    long long i = (long long)blockIdx.x * blockDim.x + threadIdx.x;
    if (i < nd) agg[i] = 0.f;
    if (i < N) { m[i] = -__builtin_huge_valf(); denom[i] = 0.f; }
}

// ---------- 4) edge logits + segment max ----------
__global__ void k_logits(const int* __restrict__ src, const int* __restrict__ dst,
                         const int* __restrict__ et,
                         const float* __restrict__ ps, const float* __restrict__ pd,
                         const float* __restrict__ pr,
                         float* __restrict__ ew, float* __restrict__ m, int E) {
    int i = blockIdx.x * blockDim.x + threadIdx.x;
    if (i >= E) return;
    int d2 = dst[i];
    float x = ps[src[i]] + pd[d2] + pr[et[i]];
    float e = x > 0.f ? x : 0.01f * x;   // leaky_relu(x, 0.01)
    ew[i] = e;
    atomicMaxFloat(&m[d2], e);
}

// ---------- 5) w = exp(e - m[dst]); denom += w ----------
__global__ void k_weights(const int* __restrict__ dst,
                          float* __restrict__ ew, const float* __restrict__ m,
                          float* __restrict__ denom, int E) {
    int i = blockIdx.x * blockDim.x + threadIdx.x;
    if (i >= E) return;
    int d2 = dst[i];
    float w = __expf(ew[i] - m[d2]);
    ew[i] = w;
    atomicAdd(&denom[d2], w);
}

// ---------- 6) weighted scatter: one wave per edge, float4 gather, atomic adds ----------
__global__ __launch_bounds__(256) void k_scatter(const int* __restrict__ src,
                                                 const int* __restrict__ dst,
                                                 const float* __restrict__ ew,
                                                 const float* __restrict__ denom,
                                                 const float* __restrict__ h,
                                                 float* __restrict__ agg, int E) {
    int edge = blockIdx.x * 8 + (threadIdx.x >> 5);
    int lane = threadIdx.x & 31;
    if (edge >= E) return;
    int s = src[edge], d2 = dst[edge];
    float alpha = ew[edge] / denom[d2];
    const float4* hv = (const float4*)(h + (size_t)s * D);
    float4 x = hv[lane];
    float* ap = agg + (size_t)d2 * D + lane * 4;
    atomicAdd(ap + 0, alpha * x.x);
    atomicAdd(ap + 1, alpha * x.y);
    atomicAdd(ap + 2, alpha * x.z);
    atomicAdd(ap + 3, alpha * x.w);
}

// ---------- 7) loopmsg = h @ loop_weight via V_WMMA_F32_16X16X4_F32, fused epilogue ----------
// Block: 256 threads = 8 waves; computes 16 rows x 128 cols. A tile staged in LDS.
// Fragment layouts (ISA 7.12.2, wave32):
//   A 16x4 f32:  a[v] = A(m = L%16, k = k0 + (L/16)*2 + v)
//   B 4x16 f32:  b[v] = B(k = k0 + (L/16)*2 + v, n = L%16)
//   C 16x16 f32: c[p] = C(m = p + (L/16)*8, n = L%16)
__global__ __launch_bounds__(256) void k_final(const float* __restrict__ h,
                                               const float* __restrict__ W,
                                               const float* __restrict__ agg,
                                               const float* __restrict__ denom,
                                               float* __restrict__ out) {
    __shared__ float As[16 * WPAD];
    int m0 = blockIdx.x * 16;
    for (int i = threadIdx.x; i < 16 * D; i += 256) {
        int r = i >> 7, c = i & (D - 1);
        As[r * WPAD + c] = h[(size_t)(m0 + r) * D + c];
    }
    __syncthreads();

    int wave = threadIdx.x >> 5;
    int L    = threadIdx.x & 31;
    int half = L >> 4;        // 0: lanes 0-15, 1: lanes 16-31
    int l16  = L & 15;
    int kgrp = half * 2;
    int n0   = wave * 16;

    v8f acc = {};
    for (int k0 = 0; k0 < D; k0 += 4) {
        v2f a, b;
        a.x = As[l16 * WPAD + k0 + kgrp];
        a.y = As[l16 * WPAD + k0 + kgrp + 1];
        b.x = W[(size_t)(k0 + kgrp)     * D + n0 + l16];
        b.y = W[(size_t)(k0 + kgrp + 1) * D + n0 + l16];
        acc = __builtin_amdgcn_wmma_f32_16x16x4_f32(
            /*neg_a=*/false, a, /*neg_b=*/false, b,
            /*c_mod=*/(short)0, acc, /*reuse_a=*/false, /*reuse_b=*/false);
    }

    for (int p = 0; p < 8; ++p) {
        int mm = m0 + p + half * 8;
        int nn = n0 + l16;
        size_t idx = (size_t)mm * D + nn;
        float base = (denom[mm] > 0.f) ? agg[idx] : h[idx];
        out[idx] = base + acc[p];
    }
}

extern "C" void kernel_launch(void* const* d_in, const int* in_sizes, int n_in,
                              void* d_out, int out_size, void* d_ws, size_t ws_size,
                              hipStream_t stream) {
    const float* h          = (const float*)d_in[0];
    const float* emb_rel    = (const float*)d_in[1];
    const float* loop_w     = (const float*)d_in[2];
    const float* attn_fc_w  = (const float*)d_in[3];
    const float* attn_fc2_w = (const float*)d_in[4];
    const int*   src        = (const int*)d_in[5];
    const int*   dst        = (const int*)d_in[6];
    const int*   etype      = (const int*)d_in[7];
    float*       out        = (float*)d_out;

    const int N = in_sizes[0] / D;   // 50000
    const int R = in_sizes[1] / D;   // 200
    const int E = in_sizes[5];       // 800000

    // workspace layout (floats)
    float* ws    = (float*)d_ws;
    float* u     = ws;                   // 3*128, padded to 512
    float* ps    = ws + 512;             // N
    float* pd    = ps + N;               // N
    float* pr    = pd + N;               // R, padded to 256
    float* m     = pr + 256;             // N
    float* denom = m + N;                // N
    float* ew    = denom + N;            // E
    float* agg   = ew + E;               // N*D

    const long long nd = (long long)N * D;

    k_reduce_u<<<1, D, 0, stream>>>(attn_fc_w, attn_fc2_w, u);

    {
        long long rows = (long long)N + R;
        int blocks = (int)((rows * 32 + 255) / 256);
        k_scalars<<<blocks, 256, 0, stream>>>(h, emb_rel, u, ps, pd, pr, N, R);
    }

    k_init<<<(int)((nd + 255) / 256), 256, 0, stream>>>(agg, m, denom, nd, N);

    k_logits<<<(E + 255) / 256, 256, 0, stream>>>(src, dst, etype, ps, pd, pr, ew, m, E);

    k_weights<<<(E + 255) / 256, 256, 0, stream>>>(dst, ew, m, denom, E);

    k_scatter<<<(E + 7) / 8, 256, 0, stream>>>(src, dst, ew, denom, h, agg, E);

    k_final<<<N / 16, 256, 0, stream>>>(h, loop_w, agg, denom, out);
}